// GPSPolicyNetwork_9225589751901
// MI455X (gfx1250) — compile-verified
//
#include <hip/hip_runtime.h>

typedef __attribute__((ext_vector_type(2))) float v2f;
typedef __attribute__((ext_vector_type(8))) float v8f;

#define NNODE 2048
#define DIN   4096
#define HIDD  512
#define NHEAD 8
#define HCH   64
#define NB    8
#define SSEQ  256
#define NEDGE 32768
#define KRW   8
#define NLAY  2
#define ETOT  (NEDGE + NNODE)
#define H4    128

#define CEILD(a,b) (((a)+(b)-1)/(b))

static __device__ __forceinline__ float geluf(float x) {
  return 0.5f * x * (1.0f + erff(x * 0.70710678118654752f));
}

static __device__ __forceinline__ void atomicMaxF(float* addr, float val) {
  int* ai = (int*)addr;
  int old = __float_as_int(*addr);
  while (__int_as_float(old) < val) {
    int assumed = old;
    old = atomicCAS(ai, assumed, __float_as_int(val));
    if (old == assumed) break;
  }
}

// ---------------------------------------------------------------------------
// Batched GEMM on V_WMMA_F32_16X16X4_F32, wave32.
// Block = 128 threads = 4 waves; each wave owns a 16-row strip and computes a
// 16x64 slab (4 N-tiles) -> block tile is 64x64.
// Register double-buffering: loads for K-step k+4 are issued before the 4
// WMMAs of K-step k execute, so load latency overlaps matrix math instead of
// the load->s_wait_loadcnt(0)->wmma serialization seen without pipelining.
// Layouts per CDNA5 ISA 7.12.2 (f32, wave32):
//   A 16x4 : lanes 0-15 hold K=0(V0),K=1(V1); lanes 16-31 hold K=2(V0),K=3(V1)
//   B 4x16 : V0 = row K=0 (lanes 0-15) / K=2 (lanes 16-31); V1 = K=1 / K=3
//   C 16x16: VGPR i -> row bm+i (lanes 0-15) / row bm+i+8 (lanes 16-31)
// All call sites have M,N % 64 == 0 and K % 4 == 0.
// ---------------------------------------------------------------------------
template <int TRANSB>
__global__ void __launch_bounds__(128)
k_wmma_gemm(const float* __restrict__ A, const float* __restrict__ B,
            float* __restrict__ C, const float* __restrict__ bias,
            int Kk, int lda, int ldb, int ldc,
            float alpha, int accum,
            int batchH, long sAb, long sAh, long sBb, long sBh, long sCb, long sCh) {
  const int lane = threadIdx.x & 31;
  const int wave = threadIdx.x >> 5;
  const int half = lane >> 4;
  const int l15  = lane & 15;
  const int bm = blockIdx.y * 64 + wave * 16;
  const int bn = blockIdx.x * 64;
  const int bz = blockIdx.z;
  const int bb = bz / batchH;
  const int hh = bz % batchH;
  const float* Ab = A + bb * sAb + hh * sAh;
  const float* Bb = B + bb * sBb + hh * sBh;
  float*       Cb = C + bb * sCb + hh * sCh;

  v8f acc[4] = {};
  const float* ap = Ab + (long)(bm + l15) * lda;

  auto load_ab = [&](int kk, v2f& a, v2f* b) {
    const int ka = kk + 2 * half;
    a[0] = ap[ka];
    a[1] = ap[ka + 1];
#pragma unroll
    for (int t = 0; t < 4; ++t) {
      const long col = bn + 16 * t + l15;
      if (TRANSB) {
        b[t][0] = Bb[col * ldb + ka];
        b[t][1] = Bb[col * ldb + ka + 1];
      } else {
        b[t][0] = Bb[(long)ka * ldb + col];
        b[t][1] = Bb[(long)(ka + 1) * ldb + col];
      }
    }
  };

  v2f a0, b0[4];
  load_ab(0, a0, b0);
  for (int k = 0; k + 4 < Kk; k += 4) {
    // issue next K-step's loads before consuming the current registers
    v2f a1, b1[4];
    load_ab(k + 4, a1, b1);
    // prefetch the streaming operands further ahead (global_prefetch_b8)
    __builtin_prefetch(ap + k + 2 * half + 64, 0, 3);
    if (!TRANSB) __builtin_prefetch(Bb + (long)(k + 2 * half + 64) * ldb + bn + l15, 0, 3);
#pragma unroll
    for (int t = 0; t < 4; ++t)
      acc[t] = __builtin_amdgcn_wmma_f32_16x16x4_f32(false, a0, false, b0[t],
                                                     (short)0, acc[t], false, false);
    a0 = a1;
#pragma unroll
    for (int t = 0; t < 4; ++t) b0[t] = b1[t];
  }
#pragma unroll
  for (int t = 0; t < 4; ++t)
    acc[t] = __builtin_amdgcn_wmma_f32_16x16x4_f32(false, a0, false, b0[t],
                                                   (short)0, acc[t], false, false);

#pragma unroll
  for (int t = 0; t < 4; ++t) {
    const long col = bn + 16 * t + l15;
    const float bv = bias ? bias[col] : 0.0f;
#pragma unroll
    for (int i = 0; i < 8; ++i) {
      const long row = bm + i + 8 * half;
      float v = alpha * acc[t][i] + bv;
      const long off = row * ldc + col;
      if (accum) v += Cb[off];
      Cb[off] = v;
    }
  }
}

// ------------------------------ small kernels ------------------------------

__global__ void k_fill(float* p, float v, long n) {
  long i = (long)blockIdx.x * blockDim.x + threadIdx.x;
  if (i < n) p[i] = v;
}

__global__ void k_scatter_adj(const int* __restrict__ ei, float* __restrict__ adj) {
  int e = blockIdx.x * blockDim.x + threadIdx.x;
  if (e < NEDGE) adj[(long)ei[e] * NNODE + ei[NEDGE + e]] = 1.0f;
}

__global__ void k_rownorm(float* __restrict__ T) {
  const int row = blockIdx.x;
  const int tid = threadIdx.x;
  __shared__ float red[256];
  float* rp = T + (long)row * NNODE;
  float s = 0.0f;
  for (int j = tid; j < NNODE; j += 256) s += rp[j];
  red[tid] = s; __syncthreads();
  for (int st = 128; st > 0; st >>= 1) { if (tid < st) red[tid] += red[tid + st]; __syncthreads(); }
  const float deg = red[0];
  const float inv = deg > 0.0f ? 1.0f / deg : 1.0f;
  for (int j = tid; j < NNODE; j += 256) rp[j] *= inv;
}

__global__ void k_diag_one(float* P) {
  int i = blockIdx.x * blockDim.x + threadIdx.x;
  if (i < NNODE) P[(long)i * NNODE + i] = 1.0f;
}

__global__ void k_diag_to_rwse(const float* __restrict__ P, float* __restrict__ rw, int step) {
  int i = blockIdx.x * blockDim.x + threadIdx.x;
  if (i < NNODE) rw[i * KRW + step] = P[(long)i * NNODE + i];
}

// y = LN(x [+resid] [+bias]) * g + b  [, gelu]
__global__ void k_layernorm(const float* __restrict__ x, const float* __restrict__ resid,
                            const float* __restrict__ bias, const float* __restrict__ g,
                            const float* __restrict__ b, float* __restrict__ y,
                            int D, int do_gelu) {
  const int row = blockIdx.x;
  const int tid = threadIdx.x;
  __shared__ float red[256];
  const float* xr = x + (long)row * D;
  const float* rr = resid ? resid + (long)row * D : nullptr;
  float s = 0.0f;
  for (int j = tid; j < D; j += 256) {
    float v = xr[j]; if (rr) v += rr[j]; if (bias) v += bias[j];
    s += v;
  }
  red[tid] = s; __syncthreads();
  for (int st = 128; st > 0; st >>= 1) { if (tid < st) red[tid] += red[tid + st]; __syncthreads(); }
  const float mu = red[0] / D;
  __syncthreads();
  float s2 = 0.0f;
  for (int j = tid; j < D; j += 256) {
    float v = xr[j]; if (rr) v += rr[j]; if (bias) v += bias[j];
    const float d = v - mu; s2 += d * d;
  }
  red[tid] = s2; __syncthreads();
  for (int st = 128; st > 0; st >>= 1) { if (tid < st) red[tid] += red[tid + st]; __syncthreads(); }
  const float inv = rsqrtf(red[0] / D + 1e-5f);
  for (int j = tid; j < D; j += 256) {
    float v = xr[j]; if (rr) v += rr[j]; if (bias) v += bias[j];
    float o = (v - mu) * inv * g[j] + b[j];
    if (do_gelu) o = geluf(o);
    y[(long)row * D + j] = o;
  }
}

__global__ void k_gelu_inplace(float* x, long n) {
  long i = (long)blockIdx.x * blockDim.x + threadIdx.x;
  if (i < n) x[i] = geluf(x[i]);
}

__global__ void k_softmax_rows(float* __restrict__ x, int cols) {
  const int row = blockIdx.x;
  const int tid = threadIdx.x;
  __shared__ float red[256];
  float* xr = x + (long)row * cols;
  float m = -3.0e38f;
  for (int j = tid; j < cols; j += 256) m = fmaxf(m, xr[j]);
  red[tid] = m; __syncthreads();
  for (int st = 128; st > 0; st >>= 1) { if (tid < st) red[tid] = fmaxf(red[tid], red[tid + st]); __syncthreads(); }
  m = red[0]; __syncthreads();
  float s = 0.0f;
  for (int j = tid; j < cols; j += 256) { const float e = expf(xr[j] - m); xr[j] = e; s += e; }
  red[tid] = s; __syncthreads();
  for (int st = 128; st > 0; st >>= 1) { if (tid < st) red[tid] += red[tid + st]; __syncthreads(); }
  const float inv = 1.0f / red[0];
  for (int j = tid; j < cols; j += 256) xr[j] *= inv;
}

__global__ void k_ea_sum(const float* __restrict__ ea, float* __restrict__ s) {
  int e = blockIdx.x * blockDim.x + threadIdx.x;
  if (e < NEDGE) {
    atomicAdd(&s[0], ea[2 * e]);
    atomicAdd(&s[1], ea[2 * e + 1]);
  }
}

static __device__ __forceinline__ void edge_sd(const int* ei, int e, int& s, int& d) {
  if (e < NEDGE) { s = ei[e]; d = ei[NEDGE + e]; }
  else           { s = e - NEDGE; d = s; }
}

__global__ void k_edge_logits(const float* __restrict__ xl, const float* __restrict__ xr,
                              const float* __restrict__ ea, const float* __restrict__ easum,
                              const float* __restrict__ We, const float* __restrict__ att,
                              const int* __restrict__ ei, float* __restrict__ logits) {
  int t = blockIdx.x * blockDim.x + threadIdx.x;
  if (t >= ETOT * NHEAD) return;
  const int e = t >> 3, hd = t & 7;
  int s, d; edge_sd(ei, e, s, d);
  float ea0, ea1;
  if (e < NEDGE) { ea0 = ea[2 * e]; ea1 = ea[2 * e + 1]; }
  else { ea0 = easum[0] / (float)NEDGE; ea1 = easum[1] / (float)NEDGE; }
  const float* xls = xl + (long)s * HIDD + hd * HCH;
  const float* xrd = xr + (long)d * HIDD + hd * HCH;
  const float* w0  = We + hd * HCH;
  const float* w1  = We + HIDD + hd * HCH;
  const float* at  = att + hd * HCH;
  float lg = 0.0f;
  for (int c = 0; c < HCH; ++c) {
    float m = xls[c] + xrd[c] + ea0 * w0[c] + ea1 * w1[c];
    m = m > 0.0f ? m : 0.2f * m;
    lg += m * at[c];
  }
  logits[t] = lg;
}

__global__ void k_seg_max(const float* __restrict__ logits, const int* __restrict__ ei,
                          float* __restrict__ smax) {
  int t = blockIdx.x * blockDim.x + threadIdx.x;
  if (t >= ETOT * NHEAD) return;
  const int e = t >> 3, hd = t & 7;
  int s, d; edge_sd(ei, e, s, d);
  atomicMaxF(&smax[d * NHEAD + hd], logits[t]);
}

__global__ void k_seg_exp(const float* __restrict__ logits, const int* __restrict__ ei,
                          const float* __restrict__ smax, float* __restrict__ ex,
                          float* __restrict__ den) {
  int t = blockIdx.x * blockDim.x + threadIdx.x;
  if (t >= ETOT * NHEAD) return;
  const int e = t >> 3, hd = t & 7;
  int s, d; edge_sd(ei, e, s, d);
  const float v = expf(logits[t] - smax[d * NHEAD + hd]);
  ex[t] = v;
  atomicAdd(&den[d * NHEAD + hd], v);
}

__global__ void k_seg_scatter(const float* __restrict__ ex, const float* __restrict__ den,
                              const int* __restrict__ ei, const float* __restrict__ xl,
                              float* __restrict__ out) {
  int t = blockIdx.x * blockDim.x + threadIdx.x;
  if (t >= ETOT * NHEAD) return;
  const int e = t >> 3, hd = t & 7;
  int s, d; edge_sd(ei, e, s, d);
  const float a = ex[t] / (den[d * NHEAD + hd] + 1e-16f);
  const float* xs = xl + (long)s * HIDD + hd * HCH;
  float* od = out + (long)d * HIDD + hd * HCH;
  for (int c = 0; c < HCH; ++c) atomicAdd(&od[c], a * xs[c]);
}

__global__ void k_combine(const float* __restrict__ local, const float* __restrict__ g,
                          const float* __restrict__ alpha_p, float* __restrict__ comb, long n) {
  long i = (long)blockIdx.x * blockDim.x + threadIdx.x;
  if (i < n) {
    const float a = 1.0f / (1.0f + expf(-alpha_p[0]));
    comb[i] = a * local[i] + (1.0f - a) * g[i];
  }
}

__global__ void k_head(const float* __restrict__ h, const int* __restrict__ idx,
                       const float* __restrict__ W1, const float* __restrict__ b1,
                       const float* __restrict__ g, const float* __restrict__ bt,
                       const float* __restrict__ W2, const float* __restrict__ b2,
                       float* __restrict__ out) {
  __shared__ float emb[HIDD], t[HIDD], red[HIDD], lg[8];
  const int tid = threadIdx.x;              // 512 threads
  const int node = idx[0];
  emb[tid] = h[(long)node * HIDD + tid];
  __syncthreads();
  float acc = 0.0f;
  for (int k = 0; k < HIDD; ++k) acc += emb[k] * W1[(long)k * HIDD + tid];
  acc += b1[tid];
  t[tid] = acc;
  red[tid] = acc; __syncthreads();
  for (int st = 256; st > 0; st >>= 1) { if (tid < st) red[tid] += red[tid + st]; __syncthreads(); }
  const float mu = red[0] / (float)HIDD;
  __syncthreads();
  const float dv = t[tid] - mu;
  red[tid] = dv * dv; __syncthreads();
  for (int st = 256; st > 0; st >>= 1) { if (tid < st) red[tid] += red[tid + st]; __syncthreads(); }
  const float inv = rsqrtf(red[0] / (float)HIDD + 1e-5f);
  __syncthreads();
  t[tid] = geluf(dv * inv * g[tid] + bt[tid]);
  __syncthreads();
  if (tid < 5) {
    float a = 0.0f;
    for (int k = 0; k < HIDD; ++k) a += t[k] * W2[k * 5 + tid];
    lg[tid] = a + b2[tid];
  }
  __syncthreads();
  if (tid == 0) {
    float mx = lg[0];
    for (int a = 1; a < 5; ++a) mx = fmaxf(mx, lg[a]);
    float sum = 0.0f, ex[5];
    for (int a = 0; a < 5; ++a) { ex[a] = expf(lg[a] - mx); sum += ex[a]; }
    for (int a = 0; a < 5; ++a) { out[a] = lg[a]; out[5 + a] = ex[a] / sum; }
  }
}

// ------------------------------ host launcher ------------------------------

extern "C" void kernel_launch(void* const* d_in, const int* in_sizes, int n_in,
                              void* d_out, int out_size, void* d_ws, size_t ws_size,
                              hipStream_t stream) {
  (void)in_sizes; (void)n_in; (void)out_size; (void)ws_size;
  const float* x        = (const float*)d_in[0];
  const float* edge_attr= (const float*)d_in[1];
  const float* rwse_W   = (const float*)d_in[2];
  const float* rwse_b   = (const float*)d_in[3];
  const float* rwse_g   = (const float*)d_in[4];
  const float* rwse_bt  = (const float*)d_in[5];
  const float* np_W     = (const float*)d_in[6];
  const float* np_b     = (const float*)d_in[7];
  const float* np_g     = (const float*)d_in[8];
  const float* np_bt    = (const float*)d_in[9];
  const float* gat_Wl   = (const float*)d_in[10];
  const float* gat_bl   = (const float*)d_in[11];
  const float* gat_Wr   = (const float*)d_in[12];
  const float* gat_br   = (const float*)d_in[13];
  const float* gat_We   = (const float*)d_in[14];
  const float* gat_att  = (const float*)d_in[15];
  const float* gat_bias = (const float*)d_in[16];
  const float* n1_g     = (const float*)d_in[17];
  const float* n1_b     = (const float*)d_in[18];
  const float* mha_inW  = (const float*)d_in[19];
  const float* mha_inb  = (const float*)d_in[20];
  const float* mha_outW = (const float*)d_in[21];
  const float* mha_outb = (const float*)d_in[22];
  const float* alpha_p  = (const float*)d_in[23];
  const float* n2_g     = (const float*)d_in[24];
  const float* n2_b     = (const float*)d_in[25];
  const float* ffn_W1   = (const float*)d_in[26];
  const float* ffn_b1   = (const float*)d_in[27];
  const float* ffn_W2   = (const float*)d_in[28];
  const float* ffn_b2   = (const float*)d_in[29];
  const float* n3_g     = (const float*)d_in[30];
  const float* n3_b     = (const float*)d_in[31];
  const float* ah_W1    = (const float*)d_in[32];
  const float* ah_b1    = (const float*)d_in[33];
  const float* ah_g     = (const float*)d_in[34];
  const float* ah_bt    = (const float*)d_in[35];
  const float* ah_W2    = (const float*)d_in[36];
  const float* ah_b2    = (const float*)d_in[37];
  const int*   edge_idx = (const int*)d_in[38];
  const int*   cur_idx  = (const int*)d_in[39];
  float* out = (float*)d_out;

  // bump allocator over workspace
  float* wp = (float*)d_ws;
  auto alloc = [&](size_t nf) { float* r = wp; wp += nf; return r; };
  float* T      = alloc((size_t)NNODE * NNODE);
  float* P0     = alloc((size_t)NNODE * NNODE);
  float* P1     = alloc((size_t)NNODE * NNODE);
  float* rwse   = alloc((size_t)NNODE * KRW);
  float* rtmp   = alloc((size_t)NNODE * H4);
  float* rbuf   = alloc((size_t)NNODE * H4);
  float* htmp   = alloc((size_t)NNODE * HIDD);
  float* h      = alloc((size_t)NNODE * HIDD);
  float* xl     = alloc((size_t)NNODE * HIDD);
  float* xr     = alloc((size_t)NNODE * HIDD);
  float* elog   = alloc((size_t)ETOT * NHEAD);
  float* eex    = alloc((size_t)ETOT * NHEAD);
  float* smax   = alloc((size_t)NNODE * NHEAD);
  float* sden   = alloc((size_t)NNODE * NHEAD);
  float* gatout = alloc((size_t)NNODE * HIDD);
  float* local  = alloc((size_t)NNODE * HIDD);
  float* qkv    = alloc((size_t)NNODE * 3 * HIDD);
  float* scores = alloc((size_t)NB * NHEAD * SSEQ * SSEQ);
  float* attno  = alloc((size_t)NNODE * HIDD);
  float* gbuf   = alloc((size_t)NNODE * HIDD);
  float* comb   = alloc((size_t)NNODE * HIDD);
  float* ffn1   = alloc((size_t)NNODE * 4 * HIDD);
  float* ffn2   = alloc((size_t)NNODE * HIDD);
  float* easum  = alloc(16);

  auto gemm = [&](const float* A, const float* B, float* C, const float* bias,
                  int M, int Nn, int Kk, int lda, int ldb, int ldc,
                  float alpha, bool transB, bool accum,
                  int batchB = 1, int batchH = 1,
                  long sAb = 0, long sAh = 0, long sBb = 0, long sBh = 0,
                  long sCb = 0, long sCh = 0) {
    dim3 grid(Nn / 64, M / 64, batchB * batchH);
    if (transB)
      k_wmma_gemm<1><<<grid, 128, 0, stream>>>(A, B, C, bias, Kk, lda, ldb, ldc,
                                               alpha, (int)accum, batchH,
                                               sAb, sAh, sBb, sBh, sCb, sCh);
    else
      k_wmma_gemm<0><<<grid, 128, 0, stream>>>(A, B, C, bias, Kk, lda, ldb, ldc,
                                               alpha, (int)accum, batchH,
                                               sAb, sAh, sBb, sBh, sCb, sCh);
  };
  auto fill = [&](float* p, float v, long n) {
    k_fill<<<CEILD(n, 256), 256, 0, stream>>>(p, v, n);
  };

  // ---- RWSE: T = row-normalized adjacency; diag(T^k), k=1..8 ----
  fill(T, 0.0f, (long)NNODE * NNODE);
  k_scatter_adj<<<CEILD(NEDGE, 256), 256, 0, stream>>>(edge_idx, T);
  k_rownorm<<<NNODE, 256, 0, stream>>>(T);
  fill(P0, 0.0f, (long)NNODE * NNODE);
  k_diag_one<<<CEILD(NNODE, 256), 256, 0, stream>>>(P0);
  float* Pc = P0; float* Pn = P1;
  for (int k = 0; k < KRW; ++k) {
    gemm(Pc, T, Pn, nullptr, NNODE, NNODE, NNODE, NNODE, NNODE, NNODE, 1.0f, false, false);
    k_diag_to_rwse<<<CEILD(NNODE, 256), 256, 0, stream>>>(Pn, rwse, k);
    float* t2 = Pc; Pc = Pn; Pn = t2;
  }
  // r = gelu(LN(rwse @ rwse_W + rwse_b))
  gemm(rwse, rwse_W, rtmp, rwse_b, NNODE, H4, KRW, KRW, H4, H4, 1.0f, false, false);
  k_layernorm<<<NNODE, 256, 0, stream>>>(rtmp, nullptr, nullptr, rwse_g, rwse_bt, rbuf, H4, 1);

  // h = gelu(LN(cat(x, r) @ np_W + np_b))  (concat realized as two GEMMs)
  gemm(x, np_W, htmp, np_b, NNODE, HIDD, DIN, DIN, HIDD, HIDD, 1.0f, false, false);
  gemm(rbuf, np_W + (long)DIN * HIDD, htmp, nullptr, NNODE, HIDD, H4, H4, HIDD, HIDD, 1.0f, false, true);
  k_layernorm<<<NNODE, 256, 0, stream>>>(htmp, nullptr, nullptr, np_g, np_bt, h, HIDD, 1);

  // mean edge_attr (self-loop fill)
  fill(easum, 0.0f, 2);
  k_ea_sum<<<CEILD(NEDGE, 256), 256, 0, stream>>>(edge_attr, easum);

  const int ETH = ETOT * NHEAD;
  for (int i = 0; i < NLAY; ++i) {
    const float* Wl = gat_Wl + (long)i * HIDD * HIDD;
    const float* Wr = gat_Wr + (long)i * HIDD * HIDD;
    // ---- GATv2 ----
    gemm(h, Wl, xl, gat_bl + i * HIDD, NNODE, HIDD, HIDD, HIDD, HIDD, HIDD, 1.0f, false, false);
    gemm(h, Wr, xr, gat_br + i * HIDD, NNODE, HIDD, HIDD, HIDD, HIDD, HIDD, 1.0f, false, false);
    k_edge_logits<<<CEILD(ETH, 256), 256, 0, stream>>>(xl, xr, edge_attr, easum,
        gat_We + i * 2 * HIDD, gat_att + i * NHEAD * HCH, edge_idx, elog);
    fill(smax, -3.0e38f, (long)NNODE * NHEAD);
    fill(sden, 0.0f, (long)NNODE * NHEAD);
    fill(gatout, 0.0f, (long)NNODE * HIDD);
    k_seg_max<<<CEILD(ETH, 256), 256, 0, stream>>>(elog, edge_idx, smax);
    k_seg_exp<<<CEILD(ETH, 256), 256, 0, stream>>>(elog, edge_idx, smax, eex, sden);
    k_seg_scatter<<<CEILD(ETH, 256), 256, 0, stream>>>(eex, sden, edge_idx, xl, gatout);
    k_layernorm<<<NNODE, 256, 0, stream>>>(gatout, nullptr, gat_bias + i * HIDD,
                                           n1_g + i * HIDD, n1_b + i * HIDD, local, HIDD, 0);
    // ---- MHA (block-diagonal per graph) ----
    gemm(h, mha_inW + (long)i * HIDD * 3 * HIDD, qkv, mha_inb + i * 3 * HIDD,
         NNODE, 3 * HIDD, HIDD, HIDD, 3 * HIDD, 3 * HIDD, 1.0f, false, false);
    // scores = Q @ K^T / sqrt(HC), batched over (b, head)
    gemm(qkv, qkv + HIDD, scores, nullptr, SSEQ, SSEQ, HCH,
         3 * HIDD, 3 * HIDD, SSEQ, 0.125f, true, false,
         NB, NHEAD, (long)SSEQ * 3 * HIDD, HCH, (long)SSEQ * 3 * HIDD, HCH,
         (long)NHEAD * SSEQ * SSEQ, (long)SSEQ * SSEQ);
    k_softmax_rows<<<NB * NHEAD * SSEQ, 256, 0, stream>>>(scores, SSEQ);
    // o = att @ V
    gemm(scores, qkv + 2 * HIDD, attno, nullptr, SSEQ, HCH, SSEQ,
         SSEQ, 3 * HIDD, HIDD, 1.0f, false, false,
         NB, NHEAD, (long)NHEAD * SSEQ * SSEQ, (long)SSEQ * SSEQ,
         (long)SSEQ * 3 * HIDD, HCH, (long)SSEQ * HIDD, HCH);
    gemm(attno, mha_outW + (long)i * HIDD * HIDD, gbuf, mha_outb + i * HIDD,
         NNODE, HIDD, HIDD, HIDD, HIDD, HIDD, 1.0f, false, false);
    k_layernorm<<<NNODE, 256, 0, stream>>>(gbuf, nullptr, nullptr,
                                           n2_g + i * HIDD, n2_b + i * HIDD, gbuf, HIDD, 0);
    // ---- gate + FFN + residual LN ----
    k_combine<<<CEILD((long)NNODE * HIDD, 256), 256, 0, stream>>>(local, gbuf, alpha_p + i,
                                                                  comb, (long)NNODE * HIDD);
    gemm(comb, ffn_W1 + (long)i * HIDD * 4 * HIDD, ffn1, ffn_b1 + i * 4 * HIDD,
         NNODE, 4 * HIDD, HIDD, HIDD, 4 * HIDD, 4 * HIDD, 1.0f, false, false);
    k_gelu_inplace<<<CEILD((long)NNODE * 4 * HIDD, 256), 256, 0, stream>>>(ffn1, (long)NNODE * 4 * HIDD);
    gemm(ffn1, ffn_W2 + (long)i * 4 * HIDD * HIDD, ffn2, ffn_b2 + i * HIDD,
         NNODE, HIDD, 4 * HIDD, 4 * HIDD, HIDD, HIDD, 1.0f, false, false);
    k_layernorm<<<NNODE, 256, 0, stream>>>(ffn2, comb, nullptr,
                                           n3_g + i * HIDD, n3_b + i * HIDD, h, HIDD, 0);
  }

  // ---- action head ----
  k_head<<<1, HIDD, 0, stream>>>(h, cur_idx, ah_W1, ah_b1, ah_g, ah_bt, ah_W2, ah_b2, out);
}